// CARMIL_39754217292334
// MI455X (gfx1250) — compile-verified
//
#include <hip/hip_runtime.h>
#include <hip/hip_bf16.h>
#include <cstdint>
#include <cstddef>

typedef float v2f __attribute__((ext_vector_type(2)));
typedef float v8f __attribute__((ext_vector_type(8)));

#define NEG_HUGE (-3.402823466e+38f)

// ---------------------------------------------------------------------------
// K1: rowdot[r] = features[r,:] . attn_w   via V_WMMA_F32_16X16X4_F32
// One wave (32 lanes) per 16-row tile. B (4x16) is attn_w chunk replicated
// across all 16 output columns, so D[m][n] == rowdot partial for every n.
// A 16x4 f32 layout: lanes 0-15 -> K{0,1}, lanes 16-31 -> K{2,3} (ISA 7.12.2).
// ---------------------------------------------------------------------------
__global__ void k1_rowdot_wmma(const float* __restrict__ F,
                               const float* __restrict__ wv,
                               float* __restrict__ rowdot,
                               int n, int d) {
  const int lane  = threadIdx.x & 31;
  const int wave  = threadIdx.x >> 5;
  const int wpb   = blockDim.x >> 5;
  const int tile  = blockIdx.x * wpb + wave;
  const int tiles = (n + 15) >> 4;
  if (tile >= tiles) return;                    // wave-uniform exit (EXEC all-1 at WMMA)
  const int row0 = tile << 4;
  const int m    = lane & 15;                   // M row within tile
  const int kh   = lane >> 4;                   // 0 -> K{0,1}, 1 -> K{2,3}
  int r = row0 + m; if (r > n - 1) r = n - 1;   // clamp (harmless, writes are guarded)
  const float* frow = F + (size_t)r * (size_t)d;

  v8f acc = {0.f, 0.f, 0.f, 0.f, 0.f, 0.f, 0.f, 0.f};
  for (int kk = 0; kk < d; kk += 4) {
    v2f a = *(const v2f*)(frow + kk + 2 * kh);  // A[m][K..K+1]
    v2f b = *(const v2f*)(wv   + kk + 2 * kh);  // B[K..K+1][n] replicated over n
    acc = __builtin_amdgcn_wmma_f32_16x16x4_f32(false, a, false, b,
                                                (short)0, acc, false, false);
  }
  // C/D 16x16 f32 layout: lane 0 col N=0 rows M=0..7 in acc[0..7];
  // lane 16 col N=0 rows M=8..15 in acc[0..7].
  if (lane == 0) {
    #pragma unroll
    for (int v = 0; v < 8; ++v) { int rr = row0 + v;     if (rr < n) rowdot[rr] = acc[v]; }
  } else if (lane == 16) {
    #pragma unroll
    for (int v = 0; v < 8; ++v) { int rr = row0 + 8 + v; if (rr < n) rowdot[rr] = acc[v]; }
  }
}

// ---------------------------------------------------------------------------
// K2: scores[i] = (window-sum of rowdot)/count + b ; per-block max -> blockmax
// ---------------------------------------------------------------------------
__global__ void k2_scores(const float* __restrict__ rowdot,
                          const float* __restrict__ attn_b,
                          const int*   __restrict__ kptr,
                          float* __restrict__ scores,
                          float* __restrict__ blockmax,
                          int n) {
  __shared__ float red[256];
  const int k = kptr[0];
  const int i = blockIdx.x * blockDim.x + threadIdx.x;
  float s = NEG_HUGE;
  if (i < n) {
    int lo = i - k; if (lo < 0) lo = 0;
    int hi = i + k; if (hi > n - 1) hi = n - 1;
    float sum = 0.f;
    for (int r = lo; r <= hi; ++r) sum += rowdot[r];  // L0/L2 cached, 9 reads
    s = sum / (float)(hi - lo + 1) + attn_b[0];
    scores[i] = s;
  }
  red[threadIdx.x] = s;
  __syncthreads();
  for (int o = blockDim.x >> 1; o > 0; o >>= 1) {
    if ((int)threadIdx.x < o)
      red[threadIdx.x] = fmaxf(red[threadIdx.x], red[threadIdx.x + o]);
    __syncthreads();
  }
  if (threadIdx.x == 0) blockmax[blockIdx.x] = red[0];
}

// ---------------------------------------------------------------------------
// K3: single block: gmax = max(blockmax) ; gsum = 0 ; bag[0..d) = 0
// ---------------------------------------------------------------------------
__global__ void k3_finalize_max(const float* __restrict__ blockmax, int nb,
                                float* __restrict__ gmax, float* __restrict__ gsum,
                                float* __restrict__ bag, int d) {
  __shared__ float red[512];
  float m = NEG_HUGE;
  for (int i = threadIdx.x; i < nb; i += blockDim.x) m = fmaxf(m, blockmax[i]);
  red[threadIdx.x] = m;
  __syncthreads();
  for (int o = blockDim.x >> 1; o > 0; o >>= 1) {
    if ((int)threadIdx.x < o)
      red[threadIdx.x] = fmaxf(red[threadIdx.x], red[threadIdx.x + o]);
    __syncthreads();
  }
  if (threadIdx.x == 0) { gmax[0] = red[0]; gsum[0] = 0.f; }
  for (int j = threadIdx.x; j < d; j += blockDim.x) bag[j] = 0.f;
}

// ---------------------------------------------------------------------------
// K4: e[i] = exp(scores[i] - gmax) in place ; atomicAdd block sums into gsum
// ---------------------------------------------------------------------------
__global__ void k4_exp(float* __restrict__ scores_e,
                       const float* __restrict__ gmax,
                       float* __restrict__ gsum, int n) {
  __shared__ float red[256];
  const int i = blockIdx.x * blockDim.x + threadIdx.x;
  float e = 0.f;
  if (i < n) {
    e = __expf(scores_e[i] - gmax[0]);
    scores_e[i] = e;
  }
  red[threadIdx.x] = e;
  __syncthreads();
  for (int o = blockDim.x >> 1; o > 0; o >>= 1) {
    if ((int)threadIdx.x < o) red[threadIdx.x] += red[threadIdx.x + o];
    __syncthreads();
  }
  if (threadIdx.x == 0) atomicAdd(gsum, red[0]);
}

// ---------------------------------------------------------------------------
// K5: w[i] = e[i]/gsum -> d_out ; u[i] = w[i]/count_i -> ws
// ---------------------------------------------------------------------------
__global__ void k5_weights(const float* __restrict__ e,
                           const float* __restrict__ gsum,
                           const int*   __restrict__ kptr,
                           float* __restrict__ wout,
                           float* __restrict__ u, int n) {
  const int i = blockIdx.x * blockDim.x + threadIdx.x;
  if (i >= n) return;
  const int k = kptr[0];
  const float inv = 1.f / gsum[0];
  const float wi = e[i] * inv;
  wout[i] = wi;
  int lo = i - k; if (lo < 0) lo = 0;
  int hi = i + k; if (hi > n - 1) hi = n - 1;
  u[i] = wi / (float)(hi - lo + 1);
}

// ---------------------------------------------------------------------------
// K6: c[r] = window-sum of u  (i in [r-k, r+k] clipped)
// ---------------------------------------------------------------------------
__global__ void k6_coeff(const float* __restrict__ u,
                         const int*   __restrict__ kptr,
                         float* __restrict__ c, int n) {
  const int r = blockIdx.x * blockDim.x + threadIdx.x;
  if (r >= n) return;
  const int k = kptr[0];
  int lo = r - k; if (lo < 0) lo = 0;
  int hi = r + k; if (hi > n - 1) hi = n - 1;
  float s = 0.f;
  for (int i = lo; i <= hi; ++i) s += u[i];
  c[r] = s;
}

// ---------------------------------------------------------------------------
// K7: bag[j] += sum_r c[r] * F[r][j]  — coalesced streaming pass with prefetch
// blockDim.x == d (512); blocks stride over rows; one f32 atomic per (blk,col)
// ---------------------------------------------------------------------------
__global__ void k7_bag(const float* __restrict__ F,
                       const float* __restrict__ c,
                       float* __restrict__ bag, int n, int d) {
  const int j = threadIdx.x;
  float acc = 0.f;
  for (int r = blockIdx.x; r < n; r += gridDim.x) {
    // speculative prefetch of the next row this block will touch
    __builtin_prefetch(F + (size_t)r * (size_t)d + (size_t)gridDim.x * (size_t)d + j, 0, 1);
    acc += c[r] * F[(size_t)r * (size_t)d + j];
  }
  atomicAdd(&bag[j], acc);
}

// ---------------------------------------------------------------------------
extern "C" void kernel_launch(void* const* d_in, const int* in_sizes, int n_in,
                              void* d_out, int out_size, void* d_ws, size_t ws_size,
                              hipStream_t stream) {
  const float* F    = (const float*)d_in[0];   // [n, d]
  const float* wv   = (const float*)d_in[1];   // [d]
  const float* bb   = (const float*)d_in[2];   // [1]
  const int*   kptr = (const int*)  d_in[3];   // [1]

  const int d = in_sizes[1];
  const int n = in_sizes[0] / d;

  float* bag  = (float*)d_out;                 // [d]   (output 0)
  float* wout = (float*)d_out + d;             // [n]   (output 1)

  // workspace layout (floats): rowdot | scores/e | u | c | blockmax | gmax | gsum
  float* ws       = (float*)d_ws;
  float* rowdot   = ws;
  float* scores   = ws + (size_t)n;
  float* u        = ws + 2 * (size_t)n;
  float* c        = ws + 3 * (size_t)n;
  const int BS    = 256;
  const int nb    = (n + BS - 1) / BS;
  float* blockmax = ws + 4 * (size_t)n;
  float* gmax     = blockmax + nb;
  float* gsum     = gmax + 1;

  // K1: wmma mat-vec, one wave per 16-row tile, 8 waves/block
  const int tiles = (n + 15) / 16;
  const int wpb   = 8;
  dim3 b1(32 * wpb), g1((tiles + wpb - 1) / wpb);
  k1_rowdot_wmma<<<g1, b1, 0, stream>>>(F, wv, rowdot, n, d);

  k2_scores      <<<nb, BS, 0, stream>>>(rowdot, bb, kptr, scores, blockmax, n);
  k3_finalize_max<<<1, 512, 0, stream>>>(blockmax, nb, gmax, gsum, bag, d);
  k4_exp         <<<nb, BS, 0, stream>>>(scores, gmax, gsum, n);
  k5_weights     <<<nb, BS, 0, stream>>>(scores, gsum, kptr, wout, u, n);
  k6_coeff       <<<nb, BS, 0, stream>>>(u, kptr, c, n);

  int g7 = 512; if (g7 > n) g7 = n;
  k7_bag<<<g7, d, 0, stream>>>(F, c, bag, n, d);
}